// CoOccurrenceSemanticGroundingLossModule_17626545783502
// MI455X (gfx1250) — compile-verified
//
#include <hip/hip_runtime.h>

typedef float v2f __attribute__((ext_vector_type(2)));
typedef float v4f __attribute__((ext_vector_type(4)));
typedef float v8f __attribute__((ext_vector_type(8)));

#define BB 128
#define TT 1024
#define VV 512
#define LL 20

// ---------------- prep: present / valid ----------------

__global__ void mark_present_kernel(const int* __restrict__ gs, int* __restrict__ present) {
    int i = blockIdx.x * blockDim.x + threadIdx.x; // over B*L
    if (i < BB * LL) {
        int b = i / LL;
        int t = gs[i];                 // token id in [0, T)
        present[b * TT + t] = 1;       // benign same-value races
    }
}

__global__ void compute_valid_kernel(const int* __restrict__ present, int* __restrict__ valid) {
    int t = blockIdx.x * blockDim.x + threadIdx.x;
    if (t < TT) {
        int cnt = 0;
        for (int b = 0; b < BB; ++b) cnt += present[b * TT + t];
        valid[t] = (cnt > 0 && cnt < BB) ? 1 : 0;
    }
}

// ---------------- WMMA wave32 all-reduce (f32) ----------------
// a[0] = per-lane partial, a[1] = 0 (32 of A's 64 entries are partials), B = all-ones:
//   D[m,n] = sum_k A[m,k]  (accumulated over groups of lanes)
// Summing the 8 C/D VGPRs gives the half-total for this lane's half;
// one shfl_xor(16) + add yields the full 32-lane sum on every lane.
// Robust to A/B lane-layout details because B is all-ones and we sum
// every D element across both lane halves.
__device__ __forceinline__ float wave_allreduce_sum_wmma(float x) {
    v2f a; a[0] = x;   a[1] = 0.0f;
    v2f bm; bm[0] = 1.0f; bm[1] = 1.0f;
    v8f c = {};
    c = __builtin_amdgcn_wmma_f32_16x16x4_f32(
        /*neg_a=*/false, a, /*neg_b=*/false, bm,
        /*c_mod=*/(short)0, c, /*reuse_a=*/false, /*reuse_b=*/false);
    float s = ((c[0] + c[1]) + (c[2] + c[3])) + ((c[4] + c[5]) + (c[6] + c[7]));
    s += __shfl_xor(s, 16, 32);
    return s;
}

// ---------------- main streaming kernel ----------------
// One wave per (b,t) row: 32 lanes x 16 floats = 512 = V.
// Coalesced non-temporal b128 loads (read-once, > L2 capacity).
__global__ void __launch_bounds__(256) row_reduce_kernel(
    const float* __restrict__ prior,       // [B,T,V]
    const float* __restrict__ logits,      // [B,T,V]
    const int*   __restrict__ present,     // [B*T]
    const int*   __restrict__ valid,       // [T]
    float*       __restrict__ rowsq,       // [B*T] ws: per-row sum of squared error
    float*       __restrict__ ent_out)     // [B*T] -> d_out + B
{
    const int wave = (blockIdx.x * blockDim.x + threadIdx.x) >> 5; // row = b*T + t
    const int lane = threadIdx.x & 31;
    const int t = wave & (TT - 1);

    const float tgt = (present[wave] && valid[t]) ? 1.0f : -1.0f;

    const v4f* prow = (const v4f*)(prior  + (size_t)wave * VV);
    const v4f* lrow = (const v4f*)(logits + (size_t)wave * VV);

    float sq = 0.0f, ent = 0.0f;
#pragma unroll
    for (int ch = 0; ch < 4; ++ch) {
        const int idx = ch * 32 + lane;                  // float4 index, coalesced
        v4f pv = __builtin_nontemporal_load(prow + idx);
        v4f lv = __builtin_nontemporal_load(lrow + idx);
#pragma unroll
        for (int j = 0; j < 4; ++j) {
            float d = lv[j] - tgt;
            sq = fmaf(d, d, sq);
            float pj = pv[j];
            ent += (pj > 0.0f) ? pj * __logf(pj) : 0.0f; // v_log_f32 path
        }
    }

    // full-wave WMMA reductions (EXEC all-ones here: no divergence yet)
    float sq_tot  = wave_allreduce_sum_wmma(sq);
    float ent_tot = wave_allreduce_sum_wmma(ent);

    if (lane == 0) {
        rowsq[wave]   = sq_tot;
        ent_out[wave] = -ent_tot;
    }
}

// ---------------- deterministic loss finalize ----------------
__global__ void finalize_loss_kernel(const float* __restrict__ rowsq, float* __restrict__ loss) {
    __shared__ float sm[256];
    const int b = blockIdx.x;
    const int tid = threadIdx.x;
    float s = 0.0f;
    for (int t = tid; t < TT; t += 256) s += rowsq[b * TT + t];
    sm[tid] = s;
    __syncthreads();
    for (int ofs = 128; ofs > 0; ofs >>= 1) {
        if (tid < ofs) sm[tid] += sm[tid + ofs];
        __syncthreads();
    }
    if (tid == 0) loss[b] = sm[0] * (1.0f / ((float)TT * (float)VV));
}

// ---------------- launch ----------------
extern "C" void kernel_launch(void* const* d_in, const int* in_sizes, int n_in,
                              void* d_out, int out_size, void* d_ws, size_t ws_size,
                              hipStream_t stream) {
    // setup_inputs order: visual(0), text(1), semantic_prior(2), semantic_prior_logits(3), grounding(4)
    const float* prior  = (const float*)d_in[2];
    const float* logits = (const float*)d_in[3];
    const int*   gs     = (const int*)d_in[4];

    // workspace layout
    int*   present = (int*)d_ws;                 // B*T ints   (512 KB)
    int*   valid   = present + BB * TT;          // T ints     (4 KB)
    float* rowsq   = (float*)(valid + TT);       // B*T floats (512 KB)

    float* loss = (float*)d_out;                 // [B]
    float* ent  = loss + BB;                     // [B*T]

    hipMemsetAsync(present, 0, (size_t)BB * TT * sizeof(int), stream);

    mark_present_kernel<<<(BB * LL + 255) / 256, 256, 0, stream>>>(gs, present);
    compute_valid_kernel<<<(TT + 255) / 256, 256, 0, stream>>>(present, valid);

    // one wave per row; 8 waves per 256-thread block
    row_reduce_kernel<<<(BB * TT) / 8, 256, 0, stream>>>(prior, logits, present, valid, rowsq, ent);

    finalize_loss_kernel<<<BB, 256, 0, stream>>>(rowsq, loss);
}